// GNN_60559038874105
// MI455X (gfx1250) — compile-verified
//
#include <hip/hip_runtime.h>
#include <hip/hip_bf16.h>

typedef float v2f __attribute__((ext_vector_type(2)));
typedef float v8f __attribute__((ext_vector_type(8)));

#define FDIM 128
#define HEADS 8
#define CHAN 16
#define NEG_SLOPE 0.2f

// ---------------- GEMM: H = X @ W  (M=N nodes, K=128, N=128) ----------------
// One wave per 16x16 output tile; 8 waves per block cover a 16x128 row strip.
// Uses V_WMMA_F32_16X16X4_F32 (f32 in, f32 acc) -- full precision for the
// attention softmax that follows.
__global__ __launch_bounds__(256) void gat_gemm_wmma(
    const float* __restrict__ X, const float* __restrict__ W,
    float* __restrict__ H, int n)
{
    const int lane = threadIdx.x & 31;
    const int wave = threadIdx.x >> 5;       // 0..7 -> 16-col tile
    const int row0 = blockIdx.x * 16;
    const int col0 = wave * 16;
    const int half = lane >> 4;              // 0: K pair {k,k+1}, 1: {k+2,k+3}
    const int lid  = lane & 15;

    int arow = row0 + lid;
    if (arow >= n) arow = n - 1;             // clamp (N multiple of 16 anyway)
    const float* __restrict__ xrow = X + (size_t)arow * FDIM;

    v8f acc = {};
    for (int k = 0; k < FDIM; k += 4) {
        const int k0 = k + 2 * half;
        v2f a, b;
        a.x = xrow[k0];
        a.y = xrow[k0 + 1];
        b.x = W[(size_t)k0 * FDIM + col0 + lid];
        b.y = W[(size_t)(k0 + 1) * FDIM + col0 + lid];
        acc = __builtin_amdgcn_wmma_f32_16x16x4_f32(
            false, a, false, b, (short)0, acc, false, false);
    }
    // D layout: VGPR r, lanes 0-15 -> M=r, lanes 16-31 -> M=8+r; N=lid.
    // Hot path: full tile -> 8 straight stores with immediate offsets.
    float* __restrict__ hp =
        H + (size_t)(row0 + 8 * half) * FDIM + col0 + lid;
    if (row0 + 16 <= n) {
        #pragma unroll
        for (int r = 0; r < 8; ++r) hp[(size_t)r * FDIM] = acc[r];
    } else {
        #pragma unroll
        for (int r = 0; r < 8; ++r)
            if (row0 + 8 * half + r < n) hp[(size_t)r * FDIM] = acc[r];
    }
}

// ---------------- per-(node,head) attention logits ----------------
__global__ __launch_bounds__(256) void gat_attn(
    const float* __restrict__ H, const float* __restrict__ asrc,
    const float* __restrict__ adst, float* __restrict__ als,
    float* __restrict__ ald, int n)
{
    const int t = blockIdx.x * blockDim.x + threadIdx.x;
    if (t >= n * HEADS) return;
    const int node = t >> 3;
    const int head = t & 7;
    const float* __restrict__ hp = H + (size_t)node * FDIM + head * CHAN;
    const float* __restrict__ as = asrc + head * CHAN;
    const float* __restrict__ ad = adst + head * CHAN;
    float s = 0.f, d = 0.f;
    #pragma unroll
    for (int c = 0; c < CHAN; ++c) {
        const float v = hp[c];
        s = fmaf(v, as[c], s);
        d = fmaf(v, ad[c], d);
    }
    als[t] = s;
    ald[t] = d;
}

// ---------------- per-layer init (acc=0, denom=0, m=enc(-inf)) --------------
__global__ __launch_bounds__(256) void gat_init(
    float* __restrict__ acc, float* __restrict__ denom,
    unsigned* __restrict__ m, int n)
{
    const int t = blockIdx.x * blockDim.x + threadIdx.x;
    if (t < n * FDIM) acc[t] = 0.f;
    if (t < n * HEADS) { denom[t] = 0.f; m[t] = 0x007FFFFFu; } // enc(-inf)
}

// monotonic unsigned encoding: float order == unsigned order
__device__ __forceinline__ unsigned enc_f(float f) {
    const unsigned u = __float_as_uint(f);
    return (u & 0x80000000u) ? ~u : (u | 0x80000000u);
}
__device__ __forceinline__ float dec_f(unsigned u) {
    return (u & 0x80000000u) ? __uint_as_float(u & 0x7FFFFFFFu)
                             : __uint_as_float(~u);
}

// ---------------- pass 1: leaky-relu scores + segment max (atomic) ----------
__global__ __launch_bounds__(256) void gat_edge_score(
    const long long* __restrict__ src, const long long* __restrict__ dst,
    const float* __restrict__ als, const float* __restrict__ ald,
    float* __restrict__ ebuf, unsigned* __restrict__ m, int nedges)
{
    const int e = blockIdx.x * blockDim.x + threadIdx.x;
    if (e >= nedges) return;
    const long long s = src[e];
    const long long d = dst[e];
    const float4* __restrict__ ap = (const float4*)(als + (size_t)s * HEADS);
    const float4* __restrict__ bp = (const float4*)(ald + (size_t)d * HEADS);
    const float4 a0 = ap[0], a1 = ap[1], b0 = bp[0], b1 = bp[1];
    float v[HEADS] = { a0.x + b0.x, a0.y + b0.y, a0.z + b0.z, a0.w + b0.w,
                       a1.x + b1.x, a1.y + b1.y, a1.z + b1.z, a1.w + b1.w };
    unsigned* __restrict__ md = m + (size_t)d * HEADS;
    #pragma unroll
    for (int hd = 0; hd < HEADS; ++hd) {
        const float x = v[hd];
        v[hd] = (x > 0.f) ? x : NEG_SLOPE * x;       // leaky_relu
        atomicMax(&md[hd], enc_f(v[hd]));
    }
    float4* __restrict__ ep = (float4*)(ebuf + (size_t)e * HEADS);
    ep[0] = make_float4(v[0], v[1], v[2], v[3]);
    ep[1] = make_float4(v[4], v[5], v[6], v[7]);
}

// ---------------- pass 2: exp(e - m[dst]) + segment sum (atomic) ------------
__global__ __launch_bounds__(256) void gat_edge_exp(
    const long long* __restrict__ dst, float* __restrict__ ebuf,
    const unsigned* __restrict__ m, float* __restrict__ denom, int nedges)
{
    const int e = blockIdx.x * blockDim.x + threadIdx.x;
    if (e >= nedges) return;
    const long long d = dst[e];
    const unsigned* __restrict__ md = m + (size_t)d * HEADS;
    float* __restrict__ dn = denom + (size_t)d * HEADS;
    float* __restrict__ ep = ebuf + (size_t)e * HEADS;
    #pragma unroll
    for (int hd = 0; hd < HEADS; ++hd) {
        const float ex = __expf(ep[hd] - dec_f(md[hd]));
        ep[hd] = ex;
        atomicAdd(&dn[hd], ex);
    }
}

// ---------------- pass 3: message aggregation (wave32 per edge) -------------
// lane owns 4 channels -> head = lane/4 ; float4 gather from h[src],
// 4x global_atomic_add_f32 into acc[dst].
__global__ __launch_bounds__(256) void gat_edge_agg(
    const float* __restrict__ H, const float* __restrict__ ebuf,
    const float* __restrict__ denom, const long long* __restrict__ src,
    const long long* __restrict__ dst, float* __restrict__ acc, int nedges)
{
    const int wavesPerBlock = blockDim.x >> 5;
    const int e = blockIdx.x * wavesPerBlock + (threadIdx.x >> 5);
    if (e >= nedges) return;
    const int lane = threadIdx.x & 31;
    const int head = lane >> 2;                       // C=16 -> 4 lanes/head
    const long long s = src[e];
    const long long d = dst[e];
    const float alpha = ebuf[(size_t)e * HEADS + head] /
                        (denom[(size_t)d * HEADS + head] + 1e-16f);
    const float4 hv = *(const float4*)(H + (size_t)s * FDIM + lane * 4);
    float* __restrict__ op = acc + (size_t)d * FDIM + lane * 4;
    atomicAdd(op + 0, hv.x * alpha);
    atomicAdd(op + 1, hv.y * alpha);
    atomicAdd(op + 2, hv.z * alpha);
    atomicAdd(op + 3, hv.w * alpha);
}

// ---------------- bias + relu + residual ----------------
// mode 0: cur = relu(acc + b)                (layer 1)
// mode 1: cur = relu(acc + b) + cur          (layers 2-4)
// mode 2: out = acc + b + cur                (layer 5, writes d_out)
__global__ __launch_bounds__(256) void gat_residual(
    const float* __restrict__ acc, const float* __restrict__ bias,
    float* __restrict__ cur, float* __restrict__ out, int mode, int total)
{
    const int t = blockIdx.x * blockDim.x + threadIdx.x;
    if (t >= total) return;
    const float v = acc[t] + bias[t & (FDIM - 1)];
    if (mode == 0)      cur[t] = fmaxf(v, 0.f);
    else if (mode == 1) cur[t] = fmaxf(v, 0.f) + cur[t];
    else                out[t] = v + cur[t];
}

extern "C" void kernel_launch(void* const* d_in, const int* in_sizes, int n_in,
                              void* d_out, int out_size, void* d_ws, size_t ws_size,
                              hipStream_t stream)
{
    const float*     x    = (const float*)d_in[0];
    const long long* ei   = (const long long*)d_in[1];   // int64 [2, E]
    const float*     Ws   = (const float*)d_in[2];       // [5,128,128]
    const float*     asrc = (const float*)d_in[3];       // [5,8,16]
    const float*     adst = (const float*)d_in[4];       // [5,8,16]
    const float*     bias = (const float*)d_in[5];       // [5,128]
    float*           outp = (float*)d_out;

    const int N = in_sizes[0] / FDIM;
    const int E = in_sizes[1] / 2;
    const long long* src = ei;
    const long long* dst = ei + E;

    // carve workspace (all chunks multiple of 16B)
    float* p = (float*)d_ws;
    float*    h    = p;                p += (size_t)N * FDIM;
    float*    acc  = p;                p += (size_t)N * FDIM;
    float*    cur  = p;                p += (size_t)N * FDIM;
    float*    als  = p;                p += (size_t)N * HEADS;
    float*    ald  = p;                p += (size_t)N * HEADS;
    unsigned* m    = (unsigned*)p;     p += (size_t)N * HEADS;
    float*    den  = p;                p += (size_t)N * HEADS;
    float*    ebuf = p;                /* E*HEADS floats */

    const int TB = 256;
    const dim3 gGemm((N + 15) / 16);
    const dim3 gElem((N * FDIM + TB - 1) / TB);
    const dim3 gAttn((N * HEADS + TB - 1) / TB);
    const dim3 gEdge((E + TB - 1) / TB);
    const dim3 gAgg((E + (TB / 32) - 1) / (TB / 32));

    for (int L = 0; L < 5; ++L) {
        const float* xin = (L == 0) ? x : cur;
        gat_gemm_wmma<<<gGemm, TB, 0, stream>>>(xin, Ws + (size_t)L * FDIM * FDIM, h, N);
        gat_init     <<<gElem, TB, 0, stream>>>(acc, den, m, N);
        gat_attn     <<<gAttn, TB, 0, stream>>>(h, asrc + L * HEADS * CHAN,
                                                adst + L * HEADS * CHAN, als, ald, N);
        gat_edge_score<<<gEdge, TB, 0, stream>>>(src, dst, als, ald, ebuf, m, E);
        gat_edge_exp  <<<gEdge, TB, 0, stream>>>(dst, ebuf, m, den, E);
        gat_edge_agg  <<<gAgg,  TB, 0, stream>>>(h, ebuf, den, src, dst, acc, E);
        const int mode = (L == 0) ? 0 : (L < 4 ? 1 : 2);
        gat_residual <<<gElem, TB, 0, stream>>>(acc, bias + L * FDIM, cur, outp, mode, N * FDIM);
    }
}